// Gradient_SC_17085379903488
// MI455X (gfx1250) — compile-verified
//
#include <hip/hip_runtime.h>

// Gradient-descent sparse coding on MI455X (gfx1250), pure-f32 via
// V_WMMA_F32_16X16X4_F32.
//
//   a0 = x @ phi
//   repeat 19x:  resid = a @ phi^T - x ;  a -= eta*(resid @ phi + s*sign(a))/256
//
// v4: single barrier per K-chunk. Pipeline per chunk:
//   s_wait_asynccnt 0  (current chunk's DMA landed)
//   barrier            (all waves' DMA landed + prev compute done -> other buf free)
//   issue async DMA for next chunk into other buf   (flies during compute)
//   compute current chunk from LDS (WMMA + ds_load only)

typedef __attribute__((ext_vector_type(2))) float v2f;
typedef __attribute__((ext_vector_type(8))) float v8f;
typedef int v4i __attribute__((vector_size(16)));
typedef __attribute__((address_space(1))) v4i gv4i;
typedef __attribute__((address_space(3))) v4i lv4i;

#define WAVES  4
#define TPB    (WAVES * 32)
#define KB     16          // K-chunk per staging round
#define APITCH 20          // A tile row pitch (floats): conflict-free
#define BNP    264         // normal-B tile row pitch:   264 % 64 == 8
#define BTP    20          // trans-B  tile row pitch:   conflict-free

#if __has_builtin(__builtin_amdgcn_global_load_async_to_lds_b128) && \
    __has_builtin(__builtin_amdgcn_s_wait_asynccnt)
#define USE_ASYNC 1
#else
#define USE_ASYNC 0
#endif

__device__ __forceinline__ void cp16(const float* g, float* l) {
#if USE_ASYNC
    __builtin_amdgcn_global_load_async_to_lds_b128(
        (gv4i*)(g), (lv4i*)(l), 0, 0);
#else
    *(float4*)l = *(const float4*)g;
#endif
}

__device__ __forceinline__ void wait_async_all() {
#if USE_ASYNC
    __builtin_amdgcn_s_wait_asynccnt(0);
#endif
}

// A tile: 16 rows x KB floats = 64 16B-chunks; wave w (lanes 0-15) takes
// chunks w*16+lane -> one EXEC-masked async instr per wave.
__device__ __forceinline__ void stageA(const float* A, float* sA,
                                       int m0, int lda, int kc,
                                       int wave, int lane) {
    if (lane < 16) {
        const int c = wave * 16 + lane;
        const int m = c >> 2, q = (c & 3) * 4;
        cp16(A + (size_t)(m0 + m) * lda + kc + q, sA + m * APITCH + q);
    }
}

// Normal B (phi): 16 k-rows x 256 n = 1024 chunks -> 8 instrs per wave.
__device__ __forceinline__ void stageB_n(const float* B, float* sB,
                                         int kc, int ldb, int nwg,
                                         int wave, int lane) {
    #pragma unroll
    for (int b = 0; b < 8; ++b) {
        const int c = b * 128 + wave * 32 + lane;
        const int r = c >> 6;            // k row   0..15
        const int q = (c & 63) * 4;      // n offset
        cp16(B + (size_t)(kc + r) * ldb + nwg + q, sB + r * BNP + q);
    }
}

// Transposed B (phi^T): 256 p-rows x 16 k = 1024 chunks -> 8 instrs per wave.
__device__ __forceinline__ void stageB_t(const float* B, float* sB,
                                         int kc, int ldb, int pwg,
                                         int wave, int lane) {
    #pragma unroll
    for (int b = 0; b < 8; ++b) {
        const int c = b * 128 + wave * 32 + lane;
        const int p = c >> 2, q = (c & 3) * 4;
        cp16(B + (size_t)(pwg + p) * ldb + kc + q, sB + p * BTP + q);
    }
}

template <int BTRANS>
__device__ __forceinline__ void compute_chunk(const float* sA, const float* sB,
                                              int lm, int lh, int wave,
                                              v8f acc[4]) {
    #pragma unroll
    for (int k = 0; k < KB; k += 4) {
        // A 16x4 frag: lanes 0-15 K={k,k+1}, lanes 16-31 K={k+2,k+3}
        const v2f af = *(const v2f*)(sA + lm * APITCH + k + 2 * lh);
        const int kk = k + 2 * lh;
        #pragma unroll
        for (int t = 0; t < 4; ++t) {
            const int cl = wave * 64 + t * 16 + lm;   // col within WG tile
            v2f bf;
            if (BTRANS) {
                bf = *(const v2f*)(sB + cl * BTP + kk);   // ds_load_b64
            } else {
                const float* bp = sB + kk * BNP + cl;     // 2x ds_load_b32
                bf.x = bp[0];
                bf.y = bp[BNP];
            }
            acc[t] = __builtin_amdgcn_wmma_f32_16x16x4_f32(
                false, af, false, bf, (short)0, acc[t], false, false);
        }
    }
}

// EPI: 0 -> C = acc ; 1 -> C = acc - X ; 2 -> C -= eta*(acc + s*sign(C))/256
template <int BTRANS, int EPI>
__global__ __launch_bounds__(TPB) void gemm_step(
    const float* __restrict__ A, const float* __restrict__ B,
    float* __restrict__ C, const float* __restrict__ X,
    const float* __restrict__ Sm,
    int K, int lda, int ldb, int ldc)
{
    __shared__ float sA[2][16 * APITCH];
    __shared__ float sB[2][BTRANS ? 256 * BTP : KB * BNP];

    const int tid  = threadIdx.x;
    const int wave = tid >> 5;
    const int lane = tid & 31;
    const int lm   = lane & 15;
    const int lh   = lane >> 4;
    const int m0   = blockIdx.y * 16;
    const int nwg  = blockIdx.x * 256;    // N window (normal) / p window (trans)

    v8f acc[4];
    #pragma unroll
    for (int t = 0; t < 4; ++t) acc[t] = 0;

    // prologue: stage chunk 0 into buffer 0
    stageA(A, &sA[0][0], m0, lda, 0, wave, lane);
    if (BTRANS) stageB_t(B, &sB[0][0], 0, ldb, nwg, wave, lane);
    else        stageB_n(B, &sB[0][0], 0, ldb, nwg, wave, lane);

    int buf = 0;
    for (int kc = 0; kc < K; kc += KB) {
        wait_async_all();                 // this wave's copies for chunk kc done
        __syncthreads();                  // all copies landed; other buf reusable
        if (kc + KB < K) {                // prefetch next chunk during compute
            stageA(A, &sA[buf ^ 1][0], m0, lda, kc + KB, wave, lane);
            if (BTRANS) stageB_t(B, &sB[buf ^ 1][0], kc + KB, ldb, nwg, wave, lane);
            else        stageB_n(B, &sB[buf ^ 1][0], kc + KB, ldb, nwg, wave, lane);
        }
        compute_chunk<BTRANS>(&sA[buf][0], &sB[buf][0], lm, lh, wave, acc);
        buf ^= 1;
    }

    // epilogue: D VGPR r -> row m0 + lh*8 + r, col = tile base + lm
    float smv = 0.0f;
    if (EPI == 2) smv = Sm[0];

    #pragma unroll
    for (int t = 0; t < 4; ++t) {
        const int col = nwg + wave * 64 + t * 16 + lm;
        #pragma unroll
        for (int r = 0; r < 8; ++r) {
            const int m = m0 + lh * 8 + r;
            const size_t off = (size_t)m * ldc + col;
            const float v = acc[t][r];
            if (EPI == 0) {
                C[off] = v;
            } else if (EPI == 1) {
                C[off] = v - X[off];
            } else {
                const float aval = C[off];
                const float sgn  = (aval > 0.0f) ? 1.0f : ((aval < 0.0f) ? -1.0f : 0.0f);
                const float da   = (v + smv * sgn) * (1.0f / 256.0f);
                C[off] = aval - 0.005f * da;
            }
        }
    }
}

extern "C" void kernel_launch(void* const* d_in, const int* in_sizes, int n_in,
                              void* d_out, int out_size, void* d_ws, size_t ws_size,
                              hipStream_t stream)
{
    (void)in_sizes; (void)n_in; (void)out_size; (void)ws_size;

    const float* x   = (const float*)d_in[0];   // [256, 3072]
    const float* phi = (const float*)d_in[1];   // [3072, 4096]
    const float* sm  = (const float*)d_in[2];   // scalar sparse_mult

    float* a     = (float*)d_out;               // [256, 4096]
    float* resid = (float*)d_ws;                // [256, 3072] scratch (3 MB)

    const dim3 blk(TPB);
    const dim3 g_init(4096 / 256, 256 / 16);
    const dim3 g_res (3072 / 256, 256 / 16);
    const dim3 g_upd (4096 / 256, 256 / 16);

    // a0 = x @ phi        (K=3072, N=4096)
    gemm_step<0, 0><<<g_init, blk, 0, stream>>>(
        x, phi, a, nullptr, nullptr, 3072, 3072, 4096, 4096);

    for (int it = 0; it < 19; ++it) {
        // resid = a @ phi^T - x   (K=4096, N=3072, B transposed)
        gemm_step<1, 1><<<g_res, blk, 0, stream>>>(
            a, phi, resid, x, nullptr, 4096, 4096, 4096, 3072);
        // a -= eta*(resid @ phi + s*sign(a))/256   (K=3072, N=4096)
        gemm_step<0, 2><<<g_upd, blk, 0, stream>>>(
            resid, phi, a, nullptr, sm, 3072, 3072, 4096, 4096);
    }
}